// Harmonic_42640435315218
// MI455X (gfx1250) — compile-verified
//
#include <hip/hip_runtime.h>

typedef float v2f __attribute__((ext_vector_type(2)));
typedef float v4f __attribute__((ext_vector_type(4)));
typedef float v8f __attribute__((ext_vector_type(8)));

// explicit global-address-space pointers so weight/output traffic lowers to
// global_load/global_store instead of flat_*.
typedef const float __attribute__((address_space(1))) * gcf;
typedef float       __attribute__((address_space(1))) * gf;

// ---- reference constants ----
constexpr float PI_F      = 3.14159265358979323846f;
constexpr float MIN_F0_C  = 20.0f / 11025.0f;
constexpr float MAX_F0_C  = 800.0f / 11025.0f;
constexpr float F0_DIFF_C = MAX_F0_C - MIN_F0_C;

// B=4, A=32 -> 128 atoms; CH=64; 4 atoms per workgroup -> 32 groups
// WMMA f32 16x16x4, N-tile = 4 atoms x 4 same-parity positions

struct NetParams {
    const float* initial_w;   // (256, 128)
    const float* initial_b;   // (256,)
    const float* layers_w[5]; // each (Cin=64, Cout=64, 4)
    const float* layers_b[5]; // each (64,)
    const float* final_w;     // (1, 64, 3)
    const float* final_b;     // (1,)
};

__device__ __forceinline__ float nlf(float v) {
    float s = (__sinf(v) + 1.0f) * 0.5f;
    return s * s;
}

// ---- LDS layout (floats). Activations: [pos][atom][ci], ci fastest, with one
// zeroed guard position before/after so fragment loads need no bounds checks.
#define OFF_BUFB 0                      // 256 guard + 128*256 + 256 guard
#define OFF_BUFA 33280                  // 256 guard + 64*256 + 256 guard
#define OFF_XT   50176                  // 16 cols x 128 latent (cols 4..15 zero)
#define OFF_YT   52224                  // 128 frames x 4 atoms
#define OFF_YT2  52736
#define ZERO_TO  52224                  // zero-fill [0, ZERO_TO) once at start
#define OFF_W    53248                  // staged A_even/A_odd: [par][co][kk], co-stride 130
#define W_COSTR  130
#define W_PARSTR (64 * W_COSTR)         // 8320
#define OFF_BL   (OFF_W + 2 * W_PARSTR) // 64 staged biases
#define LDS_FLOATS (OFF_BL + 64)        // 69952 floats = 279808 B  (< 320 KB)

// NET: 0=env 1=f0 2=harm ; NL: conv-transpose layers ; NF: output frames
template <int NL, int NF, int NET>
__global__ __launch_bounds__(256) void seqnet_kernel(
    const float* __restrict__ xin, NetParams P,
    const float* __restrict__ amp_factor,
    float* __restrict__ out_f0, float* __restrict__ out_env,
    float* __restrict__ ws_ha)
{
    extern __shared__ float smem[];
    float* bufB_g = smem + OFF_BUFB + 256;   // index by pos in [-1, 128]
    float* bufA_g = smem + OFF_BUFA + 256;   // index by pos in [-1, 64]
    float* xt     = smem + OFF_XT;
    float* yt     = smem + OFF_YT;
    float* yt2    = smem + OFF_YT2;
    float* wl     = smem + OFF_W;
    float* bl     = smem + OFF_BL;

    const int tid   = threadIdx.x;
    const int lane  = tid & 31;
    const int wave  = tid >> 5;      // 8 waves
    const int hi    = lane >> 4;     // half-wave: K / M-row split
    const int col   = lane & 15;     // N column index / A row index
    const int group = blockIdx.x;    // 32 atom groups of 4

    // zero activations (incl. guards)
    for (int i = tid * 4; i < ZERO_TO; i += 256 * 4)
        *(v4f*)(smem + i) = v4f{0.0f, 0.0f, 0.0f, 0.0f};
    __syncthreads();
    {   // x tile: cols 0..3 = atoms, cols 4..15 stay zero
        gcf gx = (gcf)xin;
        for (int i = tid; i < 512; i += 256) {
            int atom = i >> 7, lat = i & 127;
            xt[atom * 128 + lat] = gx[(group * 4 + atom) * 128 + lat];
        }
    }
    __syncthreads();

    // ---- initial 1x1 conv: (256 x 128) @ (128 x 16cols), cols 4..15 zero ----
    {
        gcf giw = (gcf)P.initial_w;
        gcf gib = (gcf)P.initial_b;
        #pragma unroll
        for (int mi = 0; mi < 2; ++mi) {
            const int mt = wave + mi * 8;
            v8f acc;
            #pragma unroll
            for (int v = 0; v < 8; ++v) acc[v] = gib[mt * 16 + v + hi * 8];
            const int arow = mt * 16 + col;
            #pragma unroll 4
            for (int ks = 0; ks < 32; ++ks) {
                int k0 = ks * 4 + hi * 2;
                v2f a, b;
                a[0] = giw[arow * 128 + k0];
                a[1] = giw[arow * 128 + k0 + 1];
                b = *(const v2f*)(xt + col * 128 + k0);
                acc = __builtin_amdgcn_wmma_f32_16x16x4_f32(false, a, false, b,
                                                            (short)0, acc, false, false);
            }
            #pragma unroll
            for (int v = 0; v < 8; ++v) {
                int row = mt * 16 + v + hi * 8;          // row = c*4 + p
                if (col < 4)                              // atom columns only
                    bufA_g[(row & 3) * 256 + col * 64 + (row >> 2)] = acc[v];
            }
        }
    }
    __syncthreads();

    // ---- ConvTranspose1d stack ----
    // out[2j]   = w[:, :, 1] x[j] + w[:, :, 3] x[j-1]
    // out[2j+1] = w[:, :, 2] x[j] + w[:, :, 0] x[j+1]
    const int jj   = col >> 2;
    const int atom = col & 3;
    #pragma unroll
    for (int l = 0; l < NL; ++l) {
        const int L = 4 << l;
        float* bin  = (l & 1) ? bufB_g : bufA_g;
        float* bout = (l & 1) ? bufA_g : bufB_g;

        // stage A_even / A_odd (64 co x 128 kk each) + bias into LDS
        {
            gcf gw = (gcf)P.layers_w[l];            // constant index after unroll
            gcf gb = (gcf)P.layers_b[l];
            for (int i = tid; i < 16384; i += 256) {
                int par = i >> 13;
                int rem = i & 8191;
                int co  = rem >> 7;
                int kk  = rem & 127;
                int ci  = kk & 63;
                int tap = kk >> 6;
                int wk  = par ? (tap ? 0 : 2) : (tap ? 3 : 1);
                wl[par * W_PARSTR + co * W_COSTR + kk] = gw[ci * 256 + co * 4 + wk];
            }
            if (tid < 64) bl[tid] = gb[tid];
        }
        __syncthreads();

        const int ntiles = L >> 2;
        const int units  = 2 * ntiles * 4;   // parity x jtile x mtile
        for (int u = wave; u < units; u += 8) {
            int mt   = u & 3;
            int rest = u >> 2;
            int jt   = rest & (ntiles - 1);
            int par  = rest >> (l);          // rest / ntiles, ntiles = 2^l
            int j    = jt * 4 + jj;
            v8f acc;
            #pragma unroll
            for (int v = 0; v < 8; ++v) acc[v] = bl[mt * 16 + v + hi * 8];
            const float* wrow = wl + par * W_PARSTR + (mt * 16 + col) * W_COSTR;
            const int joff = par ? 1 : -1;   // tap-1 position offset
            #pragma unroll 4
            for (int ks = 0; ks < 32; ++ks) {
                int kk0 = ks * 4 + hi * 2;          // K-pair (same tap)
                int ci0 = kk0 & 63;
                int tap = kk0 >> 6;
                int jpos = j + (tap ? joff : 0);    // guard rows make this safe
                v2f a = *(const v2f*)(wrow + kk0);
                v2f b = *(const v2f*)(bin + jpos * 256 + atom * 64 + ci0);
                acc = __builtin_amdgcn_wmma_f32_16x16x4_f32(false, a, false, b,
                                                            (short)0, acc, false, false);
            }
            int t = 2 * j + par;
            #pragma unroll
            for (int v = 0; v < 8; ++v) {
                float val = acc[v];
                acc[v] = (val >= 0.0f) ? val : 0.2f * val;   // LeakyReLU(0.2)
            }
            float* dst = bout + t * 256 + atom * 64 + mt * 16 + hi * 8;
            *(v4f*)(dst)     = v4f{acc[0], acc[1], acc[2], acc[3]};
            *(v4f*)(dst + 4) = v4f{acc[4], acc[5], acc[6], acc[7]};
        }
        __syncthreads();
    }

    // ---- final Conv1d(64->1, k=3, pad=1) ----
    float* hfin = (NL & 1) ? bufB_g : bufA_g;
    gcf gfw = (gcf)P.final_w;
    const float fb = ((gcf)P.final_b)[0];
    for (int i = tid; i < 4 * NF; i += 256) {
        int at = i & 3, t = i >> 2;
        float acc = fb;
        for (int ci = 0; ci < 64; ++ci) {
            float w0 = gfw[ci * 3 + 0];
            float w1 = gfw[ci * 3 + 1];
            float w2 = gfw[ci * 3 + 2];
            if (t > 0)      acc += w0 * hfin[(t - 1) * 256 + at * 64 + ci];
                            acc += w1 * hfin[t * 256 + at * 64 + ci];
            if (t < NF - 1) acc += w2 * hfin[(t + 1) * 256 + at * 64 + ci];
        }
        yt[i] = acc;
    }
    __syncthreads();

    const int gbase = group * 4;
    if (NET == 0) {                       // env = nl(y)*amp
        float amp = ((gcf)amp_factor)[0];
        gf oenv = (gf)out_env;
        for (int i = tid; i < 4 * NF; i += 256) {
            int at = i & 3, t = i >> 2;
            oenv[(gbase + at) * 128 + t] = nlf(yt[i]) * amp;
        }
    } else if (NET == 1) {                // f0 = smooth(MIN + nl(y)*DIFF)
        for (int i = tid; i < 4 * NF; i += 256)
            yt2[i] = MIN_F0_C + nlf(yt[i]) * F0_DIFF_C;
        __syncthreads();
        gf of0 = (gf)out_f0;
        for (int i = tid; i < 4 * NF; i += 256) {
            int at = i & 3, t = i >> 2;
            int tm = (t == 0) ? 1 : t - 1;           // reflect pad
            int tp = (t == NF - 1) ? NF - 2 : t + 1;
            of0[(gbase + at) * 128 + t] =
                (yt2[tm * 4 + at] + yt2[i] + yt2[tp * 4 + at]) * (1.0f / 3.0f);
        }
    } else {                              // 8 harmonic gains per atom -> scratch
        gf oha = (gf)ws_ha;
        for (int i = tid; i < 4 * NF; i += 256) {
            int at = i & 3, t = i >> 2;
            oha[(gbase + at) * 8 + t] = nlf(yt[i]);
        }
    }
}

// ---- synthesis: phase = pi*cumsum(f0_u); out = env_u*(sin(p) + sum ha[h] sin((h+2)p))
__global__ __launch_bounds__(256) void synth_kernel(
    const float* __restrict__ f0_all, const float* __restrict__ env_all,
    const float* __restrict__ ha_all, float* __restrict__ out)
{
    __shared__ float f0s[128], envs[128], has[8], scan[256];
    const int g = blockIdx.x, tid = threadIdx.x;
    {
        gcf gf0 = (gcf)f0_all; gcf gev = (gcf)env_all; gcf gha = (gcf)ha_all;
        if (tid < 128) { f0s[tid] = gf0[g * 128 + tid]; envs[tid] = gev[g * 128 + tid]; }
        if (tid < 8)   has[tid] = gha[g * 8 + tid];
    }
    __syncthreads();

    const float inv = 1.0f / 256.0f;     // 128 frames -> 32768 samples
    const int base = tid << 7;           // 128 samples per thread

    float s = 0.0f;                      // pass 1: segment sum of f0_u
    for (int i = 0; i < 128; ++i) {
        int n = base + i;
        float c = fminf(fmaxf((n + 0.5f) * inv - 0.5f, 0.0f), 127.0f);
        int i0 = (int)c; int i1 = min(i0 + 1, 127); float w = c - (float)i0;
        s += f0s[i0] + w * (f0s[i1] - f0s[i0]);
    }
    scan[tid] = s;
    __syncthreads();
    float pre = 0.0f;                    // uniform-trip predicated prefix
    for (int t = 0; t < 256; ++t) pre += (t < tid) ? scan[t] : 0.0f;

    float phase = pre * PI_F;
    gf dst = (gf)out + (size_t)g * 32768;
    for (int i = 0; i < 128; ++i) {      // pass 2: oscillators
        int n = base + i;
        float c = fminf(fmaxf((n + 0.5f) * inv - 0.5f, 0.0f), 127.0f);
        int i0 = (int)c; int i1 = min(i0 + 1, 127); float w = c - (float)i0;
        float f0u  = f0s[i0]  + w * (f0s[i1]  - f0s[i0]);
        float envu = envs[i0] + w * (envs[i1] - envs[i0]);
        phase += f0u * PI_F;
        float acc = __sinf(phase);
        #pragma unroll
        for (int h = 0; h < 8; ++h) acc += has[h] * __sinf((float)(h + 2) * phase);
        dst[n] = envu * acc;
    }
}

extern "C" void kernel_launch(void* const* d_in, const int* in_sizes, int n_in,
                              void* d_out, int out_size, void* d_ws, size_t ws_size,
                              hipStream_t stream) {
    (void)in_sizes; (void)n_in; (void)out_size; (void)ws_size;
    // d_in flattening follows setup_inputs() insertion order:
    // [0] x | per net: initial_w, initial_b, layers_w[0..n-1], layers_b[0..n-1],
    // final_w, final_b | env(1..14), f0(15..28), harm(29..34) | [35] amp_factor
    const float* x = (const float*)d_in[0];
    auto mknet = [&](int base, int nl) {
        NetParams p{};
        p.initial_w = (const float*)d_in[base + 0];
        p.initial_b = (const float*)d_in[base + 1];
        for (int i = 0; i < nl; ++i) p.layers_w[i] = (const float*)d_in[base + 2 + i];
        for (int i = 0; i < nl; ++i) p.layers_b[i] = (const float*)d_in[base + 2 + nl + i];
        p.final_w = (const float*)d_in[base + 2 + 2 * nl];
        p.final_b = (const float*)d_in[base + 3 + 2 * nl];
        return p;
    };
    NetParams penv = mknet(1, 5);
    NetParams pf0  = mknet(15, 5);
    NetParams pha  = mknet(29, 1);
    const float* amp = (const float*)d_in[35];

    float* out     = (float*)d_out;            // (4,32,32768,1)
    float* out_f0  = out + 4194304;            // (4,32,128,1) smoothed f0
    float* out_env = out + 4194304 + 16384;    // (4,32,128,1) env
    float* ws_ha   = (float*)d_ws;             // 128 atoms * 8 harmonic gains

    size_t lds = LDS_FLOATS * sizeof(float);
    seqnet_kernel<5, 128, 0><<<dim3(32), 256, lds, stream>>>(x, penv, amp,
                                                             out_f0, out_env, ws_ha);
    seqnet_kernel<5, 128, 1><<<dim3(32), 256, lds, stream>>>(x, pf0, amp,
                                                             out_f0, out_env, ws_ha);
    seqnet_kernel<1, 8, 2><<<dim3(32), 256, lds, stream>>>(x, pha, amp,
                                                           out_f0, out_env, ws_ha);
    synth_kernel<<<dim3(128), 256, 0, stream>>>(out_f0, out_env, ws_ha, out);
}